// UniGATConv_47278999994500
// MI455X (gfx1250) — compile-verified
//
#include <hip/hip_runtime.h>
#include <math.h>

typedef float v2f __attribute__((ext_vector_type(2)));
typedef float v8f __attribute__((ext_vector_type(8)));

#define N_NODES 100000
#define N_EDGES 20000
#define N_PAIRS 640000
#define IN_CH   128
#define HEADS   4
#define OUT_CH  32
#define HC      (HEADS * OUT_CH)   // 128

// ---------- monotone float <-> uint encoding for atomic max over floats ----------
__device__ __forceinline__ unsigned enc_f32(float f) {
    int i = __float_as_int(f);
    return (i >= 0) ? ((unsigned)i ^ 0x80000000u) : ~(unsigned)i;
}
__device__ __forceinline__ float dec_f32(unsigned u) {
    int i = (u & 0x80000000u) ? (int)(u ^ 0x80000000u) : ~(int)u;
    return __int_as_float(i);
}

// ---------- K1: X0 = X @ W via V_WMMA_F32_16X16X4_F32 ----------
// One wave computes one 16x16 tile of X0. Block = 256 threads = 8 waves = 8 col
// tiles => each block covers 16 rows x 128 cols. Grid = 100000/16 = 6250 blocks.
__global__ __launch_bounds__(256)
void gemm_wmma_f32(const float* __restrict__ X, const float* __restrict__ W,
                   float* __restrict__ X0) {
    const int lane  = threadIdx.x & 31;
    const int wave  = threadIdx.x >> 5;
    const int row0  = blockIdx.x * 16;
    const int col0  = wave * 16;
    const int mlane = lane & 15;          // M (for A) / N (for B,C,D) within tile
    const int khalf = (lane >> 4) * 2;    // lanes 16..31 hold K+2,K+3

    v8f c = {0.f, 0.f, 0.f, 0.f, 0.f, 0.f, 0.f, 0.f};
    const float* Arow = X + (size_t)(row0 + mlane) * IN_CH;

    #pragma unroll 4
    for (int k = 0; k < IN_CH; k += 4) {
        v2f a, b;
        // A 16x4 f32: lane L holds M=L%16; VGPR0/1 = K = khalf, khalf+1
        a.x = Arow[k + khalf + 0];
        a.y = Arow[k + khalf + 1];
        // B 4x16 f32: lane L holds N=L%16; VGPR0/1 = rows K = khalf, khalf+1
        b.x = W[(size_t)(k + khalf + 0) * HC + col0 + mlane];
        b.y = W[(size_t)(k + khalf + 1) * HC + col0 + mlane];
        c = __builtin_amdgcn_wmma_f32_16x16x4_f32(
                /*neg_a=*/false, a, /*neg_b=*/false, b,
                /*c_mod=*/(short)0, c, /*reuse_a=*/false, /*reuse_b=*/false);
    }

    // C/D 16x16 f32 layout: VGPR v holds M=v (lanes 0-15) and M=v+8 (lanes 16-31)
    const int nout  = col0 + mlane;
    const int mbase = row0 + ((lane >> 4) << 3);
    #pragma unroll
    for (int v = 0; v < 8; ++v)
        X0[(size_t)(mbase + v) * HC + nout] = c[v];
}

// ---------- K2: per-pair scatter of X0 rows into hyperedge sums ----------
__global__ __launch_bounds__(128)
void scatter_edge_sum(const float* __restrict__ X0, const int* __restrict__ vertex,
                      const int* __restrict__ edges, float* __restrict__ esum,
                      float* __restrict__ ecnt) {
    const int m = blockIdx.x;
    const int t = threadIdx.x;
    const int v = vertex[m];
    const int e = edges[m];
    atomicAdd(&esum[(size_t)e * HC + t], X0[(size_t)v * HC + t]);
    if (t == 0) atomicAdd(&ecnt[e], 1.0f);
}

// ---------- K3: Xe = esum / max(cnt,1) (in place); alpha_e = sum_c Xe*att_e ----------
// Block = 128 threads = 4 wave32s; each wave owns one head -> shuffle reduction.
__global__ __launch_bounds__(128)
void edge_mean_attn(float* __restrict__ esum /* becomes Xe */,
                    const float* __restrict__ ecnt,
                    const float* __restrict__ att_e,
                    float* __restrict__ alpha_e) {
    const int e = blockIdx.x;
    const int t = threadIdx.x;       // t = h*32 + c
    const int h = t >> 5;
    const int c = t & 31;
    const float cnt = fmaxf(ecnt[e], 1.0f);
    const float xe  = esum[(size_t)e * HC + t] / cnt;
    esum[(size_t)e * HC + t] = xe;   // store Xe in place
    float p = xe * att_e[t];         // att_e is [1,H,C] flat = [h*32+c]
    #pragma unroll
    for (int off = 16; off > 0; off >>= 1)
        p += __shfl_xor(p, off, 32);
    if (c == 0) alpha_e[e * HEADS + h] = p;
}

// ---------- K4: leaky-relu per (pair, head) + segment max via atomic max ----------
__global__ void pair_lrelu_max(const float* __restrict__ alpha_e,
                               const int* __restrict__ vertex,
                               const int* __restrict__ edges,
                               float* __restrict__ alpha_pair,
                               unsigned* __restrict__ amax_u) {
    const int i = blockIdx.x * blockDim.x + threadIdx.x;
    if (i >= N_PAIRS * HEADS) return;
    const int m = i >> 2, h = i & 3;
    float a = alpha_e[edges[m] * HEADS + h];
    a = (a > 0.f) ? a : 0.01f * a;
    alpha_pair[i] = a;
    atomicMax(&amax_u[vertex[m] * HEADS + h], enc_f32(a));
}

// ---------- K5: exp(alpha - amax) + segment sum ----------
__global__ void pair_exp_sum(const int* __restrict__ vertex,
                             float* __restrict__ alpha_pair,
                             const unsigned* __restrict__ amax_u,
                             float* __restrict__ asum) {
    const int i = blockIdx.x * blockDim.x + threadIdx.x;
    if (i >= N_PAIRS * HEADS) return;
    const int m = i >> 2, h = i & 3;
    const int v = vertex[m];
    const float amax = dec_f32(amax_u[v * HEADS + h]);
    const float a = __expf(alpha_pair[i] - amax);
    alpha_pair[i] = a;
    atomicAdd(&asum[v * HEADS + h], a);
}

// ---------- K6: out[v] += Xe[e] * (alpha / (asum + eps));  out already = X0 ----------
__global__ __launch_bounds__(128)
void scatter_out(const float* __restrict__ Xe, const int* __restrict__ vertex,
                 const int* __restrict__ edges, const float* __restrict__ alpha_pair,
                 const float* __restrict__ asum, float* __restrict__ out) {
    const int m = blockIdx.x;
    const int t = threadIdx.x;
    const int h = t >> 5;
    const int v = vertex[m];
    const int e = edges[m];
    const float w = alpha_pair[m * HEADS + h] / (asum[v * HEADS + h] + 1e-8f);
    atomicAdd(&out[(size_t)v * HC + t], Xe[(size_t)e * HC + t] * w);
}

extern "C" void kernel_launch(void* const* d_in, const int* in_sizes, int n_in,
                              void* d_out, int out_size, void* d_ws, size_t ws_size,
                              hipStream_t stream) {
    (void)in_sizes; (void)n_in; (void)out_size; (void)ws_size;

    const float* X      = (const float*)d_in[0];
    const float* W      = (const float*)d_in[1];
    const float* att_e  = (const float*)d_in[2];
    const int*   vertex = (const int*)d_in[3];
    const int*   edges  = (const int*)d_in[4];
    float*       out    = (float*)d_out;

    // Workspace layout (accumulators first so one memset covers them).
    char* p = (char*)d_ws;
    float*    esum       = (float*)p;    p += sizeof(float)    * (size_t)N_EDGES * HC;     // 10.24 MB, becomes Xe
    float*    ecnt       = (float*)p;    p += sizeof(float)    * (size_t)N_EDGES;
    unsigned* amax_u     = (unsigned*)p; p += sizeof(unsigned) * (size_t)N_NODES * HEADS;  // enc(x): 0 == -inf sentinel
    float*    asum       = (float*)p;    p += sizeof(float)    * (size_t)N_NODES * HEADS;
    const size_t accum_bytes = (size_t)(p - (char*)d_ws);
    float*    alpha_e    = (float*)p;    p += sizeof(float)    * (size_t)N_EDGES * HEADS;  // fully overwritten
    float*    alpha_pair = (float*)p;    p += sizeof(float)    * (size_t)N_PAIRS * HEADS;  // fully overwritten

    hipMemsetAsync(d_ws, 0, accum_bytes, stream);

    // X0 = X @ W  -> written straight into d_out (serves as gather source and
    // as the "+ X0" residual that K6 accumulates onto).
    gemm_wmma_f32<<<N_NODES / 16, 256, 0, stream>>>(X, W, out);

    scatter_edge_sum<<<N_PAIRS, 128, 0, stream>>>(out, vertex, edges, esum, ecnt);

    edge_mean_attn<<<N_EDGES, 128, 0, stream>>>(esum, ecnt, att_e, alpha_e);

    const int tot = N_PAIRS * HEADS;
    pair_lrelu_max<<<(tot + 255) / 256, 256, 0, stream>>>(alpha_e, vertex, edges,
                                                          alpha_pair, amax_u);
    pair_exp_sum<<<(tot + 255) / 256, 256, 0, stream>>>(vertex, alpha_pair,
                                                        amax_u, asum);

    scatter_out<<<N_PAIRS, 128, 0, stream>>>(esum /* = Xe */, vertex, edges,
                                             alpha_pair, asum, out);
}